// DisenGCN_81226421502640
// MI455X (gfx1250) — compile-verified
//
#include <hip/hip_runtime.h>
#include <math.h>

// Problem constants (match reference)
#define NNODES 50000
#define NEDGES 800000
#define INPD   256
#define HID    96       // K*FAC
#define KCAP   8
#define FACD   12
#define ROUTIT 6
#define NLAYERS 4
#define EPSV   1e-12f

typedef __attribute__((ext_vector_type(2))) float v2f;
typedef __attribute__((ext_vector_type(8))) float v8f;

// ---------------------------------------------------------------------------
// Repack W_init [K=8][256][12] -> Wc [256][96] with Wc[i][k*12+o] = W[k][i][o]
// ---------------------------------------------------------------------------
__global__ void repack_w(const float* __restrict__ W, float* __restrict__ Wc) {
  int t = blockIdx.x * blockDim.x + threadIdx.x;
  if (t >= INPD * HID) return;
  int i  = t / HID;
  int ko = t - i * HID;
  int k  = ko / FACD;
  int o  = ko - k * FACD;
  Wc[t] = W[(size_t)k * INPD * FACD + (size_t)i * FACD + o];
}

// ---------------------------------------------------------------------------
// Init GEMM via V_WMMA_F32_16X16X4_F32:
//   out[N,96] = relu(X[N,256] @ Wc[256,96] + b)
// Block = 128 threads (4 waves), all waves share one 16-wide column tile.
// The 256x16 B tile (16 KB) is staged into LDS ONCE per block via the CDNA5
// async global->LDS path (global_load_async_to_lds_b128, ASYNCcnt), then each
// wave's B fragments come from LDS (ds_load) -- 4x reuse, no VGPR staging.
//
// Fragment layouts (ISA 7.12.2, 32-bit):
//   A (16x4): lanes 0-15 row M=lane K={k0,k0+1}; lanes 16-31 K={k0+2,k0+3}
//   B (4x16): lanes 0-15 col N=lane K={k0,k0+1}; lanes 16-31 K={k0+2,k0+3}
//   C/D: VGPR r <- (M=r, N=lane) lane<16 ; (M=8+r, N=lane-16) lane>=16
// ---------------------------------------------------------------------------
__global__ void init_gemm_wmma(const float* __restrict__ X,
                               const float* __restrict__ Wc,
                               const float* __restrict__ bias,
                               float* __restrict__ out) {
  __shared__ __align__(16) float lB[INPD * 16];   // lB[k*16 + n] = Wc[k][col0+n]

  const int lane = threadIdx.x & 31;
  const int wid  = threadIdx.x >> 5;
  const int rowTile = blockIdx.x * 4 + wid;       // wave-uniform
  const int colTile = blockIdx.y;                 // 0..5
  const int col0 = colTile * 16;

  // ---- Async-fill LDS with the B tile: 1024 x 16B chunks, 8 per thread ----
  {
    const unsigned ldsBase = (unsigned)(size_t)&lB[0];
    const float* wcCol = Wc + col0;
#pragma unroll
    for (int i = 0; i < 8; ++i) {
      int q  = threadIdx.x + i * 128;             // chunk id 0..1023
      int k  = q >> 2;                            // K row 0..255
      int ng = q & 3;                             // 4-float group in row
      unsigned lo = ldsBase + (unsigned)(k * 16 + ng * 4) * 4u;
      unsigned long long ga =
          (unsigned long long)(size_t)(wcCol + (size_t)k * HID + ng * 4);
      asm volatile("global_load_async_to_lds_b128 %0, %1, off"
                   :: "v"(lo), "v"(ga)
                   : "memory");
    }
    asm volatile("s_wait_asynccnt 0" ::: "memory");
  }
  __syncthreads();   // fill visible to all 4 waves; executed by ALL threads

  if (rowTile * 16 < NNODES) {                    // wave-uniform guard
    const int row0 = rowTile * 16;
    const int lh = lane >> 4;                     // lane half -> K pair
    const int lm = lane & 15;

    const float* aRow = X + (size_t)(row0 + lm) * INPD + 2 * lh;
    const float* lbp  = lB + 2 * lh * 16 + lm;    // LDS B fragment base

    v8f acc = {};
#pragma unroll 8
    for (int k0 = 0; k0 < INPD; k0 += 4) {
      v2f a, b;
      a.x = aRow[k0];
      a.y = aRow[k0 + 1];
      b.x = lbp[k0 * 16];          // ds_load: (K=k0+2lh,   N=lm)
      b.y = lbp[k0 * 16 + 16];     // ds_load: (K=k0+2lh+1, N=lm)
      acc = __builtin_amdgcn_wmma_f32_16x16x4_f32(
          /*neg_a=*/false, a, /*neg_b=*/false, b,
          /*c_mod=*/(short)0, acc, /*reuse_a=*/false, /*reuse_b=*/false);
    }

    const float bv = bias[col0 + lm];
#pragma unroll
    for (int r = 0; r < 8; ++r) {
      int m = r + 8 * lh;
      float v = acc[r] + bv;
      out[(size_t)(row0 + m) * HID + col0 + lm] = fmaxf(v, 0.0f);
    }
  }
}

// ---------------------------------------------------------------------------
// z = l2norm(relu(src)) per capsule; also initializes c = z.
// One thread per (node, capsule). Safe with src == c (thread-local RMW).
// ---------------------------------------------------------------------------
__global__ void normalize_zc(const float* __restrict__ src,
                             float* __restrict__ z,
                             float* __restrict__ c) {
  int t = blockIdx.x * blockDim.x + threadIdx.x;
  if (t >= NNODES * KCAP) return;
  const size_t base = (size_t)(t >> 3) * HID + (size_t)(t & 7) * FACD;
  float v[FACD];
  float ss = 0.0f;
#pragma unroll
  for (int o = 0; o < FACD; ++o) {
    float x = fmaxf(src[base + o], 0.0f);
    v[o] = x;
    ss += x * x;
  }
  float inv = __frsqrt_rn(ss + EPSV);
#pragma unroll
  for (int o = 0; o < FACD; ++o) {
    float y = v[o] * inv;
    z[base + o] = y;
    c[base + o] = y;
  }
}

// ---------------------------------------------------------------------------
// Per-edge routing step: att[k] = z[src]·c[dst] per capsule, softmax over K,
// scatter-add p[k]*z[src] into agg[dst]. Thread per edge, 128-bit loads,
// non-returning global_atomic_add_f32 (STOREcnt path) for the scatter.
// Working set (z,c,agg = 58 MB) is L2-resident on MI455X (192 MB L2).
// ---------------------------------------------------------------------------
__global__ void edge_route(const int* __restrict__ src,
                           const int* __restrict__ dst,
                           const float* __restrict__ z,
                           const float* __restrict__ c,
                           float* __restrict__ agg) {
  int e = blockIdx.x * blockDim.x + threadIdx.x;
  if (e >= NEDGES) return;
  const int s = src[e];
  const int d = dst[e];
  const float4* zs = (const float4*)(z + (size_t)s * HID);   // 24 x float4
  const float4* cd = (const float4*)(c + (size_t)d * HID);

  float zl[HID];
  float att[KCAP];
#pragma unroll
  for (int k = 0; k < KCAP; ++k) {
    float acc = 0.0f;
#pragma unroll
    for (int q = 0; q < 3; ++q) {
      float4 zv = zs[k * 3 + q];
      float4 cv = cd[k * 3 + q];
      int o = k * FACD + q * 4;
      zl[o + 0] = zv.x; zl[o + 1] = zv.y; zl[o + 2] = zv.z; zl[o + 3] = zv.w;
      acc += zv.x * cv.x + zv.y * cv.y + zv.z * cv.z + zv.w * cv.w;
    }
    att[k] = acc;   // TAU == 1.0
  }

  float mx = att[0];
#pragma unroll
  for (int k = 1; k < KCAP; ++k) mx = fmaxf(mx, att[k]);
  float den = 0.0f;
#pragma unroll
  for (int k = 0; k < KCAP; ++k) {
    float p = __expf(att[k] - mx);
    att[k] = p;
    den += p;
  }
  const float inv = 1.0f / den;

  float* ad = agg + (size_t)d * HID;
#pragma unroll
  for (int k = 0; k < KCAP; ++k) {
    float p = att[k] * inv;
#pragma unroll
    for (int o = 0; o < FACD; ++o)
      atomicAdd(ad + k * FACD + o, p * zl[k * FACD + o]);
  }
}

// ---------------------------------------------------------------------------
// c = l2norm(z + agg) per capsule, and re-zero agg for the next iteration
// (folds the zero pass into this kernel: saves one 19.2 MB store pass and a
// launch per routing iteration). One thread per (node, capsule).
// ---------------------------------------------------------------------------
__global__ void center_update(const float* __restrict__ z,
                              float* __restrict__ agg,
                              float* __restrict__ c) {
  int t = blockIdx.x * blockDim.x + threadIdx.x;
  if (t >= NNODES * KCAP) return;
  const size_t base = (size_t)(t >> 3) * HID + (size_t)(t & 7) * FACD;
  float v[FACD];
  float ss = 0.0f;
#pragma unroll
  for (int o = 0; o < FACD; ++o) {
    float x = z[base + o] + agg[base + o];
    v[o] = x;
    ss += x * x;
  }
#pragma unroll
  for (int o = 0; o < FACD; ++o) agg[base + o] = 0.0f;   // ready for next iter
  float inv = __frsqrt_rn(ss + EPSV);
#pragma unroll
  for (int o = 0; o < FACD; ++o) c[base + o] = v[o] * inv;
}

// ---------------------------------------------------------------------------
__global__ void zero_buf4(float4* __restrict__ p, int n4) {
  int t = blockIdx.x * blockDim.x + threadIdx.x;
  if (t < n4) p[t] = make_float4(0.f, 0.f, 0.f, 0.f);
}

__global__ void relu_out(const float* __restrict__ c, float* __restrict__ out, int n) {
  int t = blockIdx.x * blockDim.x + threadIdx.x;
  if (t < n) out[t] = fmaxf(c[t], 0.0f);
}

// ---------------------------------------------------------------------------
extern "C" void kernel_launch(void* const* d_in, const int* in_sizes, int n_in,
                              void* d_out, int out_size, void* d_ws, size_t ws_size,
                              hipStream_t stream) {
  const float* X     = (const float*)d_in[0];   // [N,256]
  const int*   edges = (const int*)d_in[1];     // [2,M]
  const float* W     = (const float*)d_in[2];   // [8,256,12]
  const float* b     = (const float*)d_in[3];   // [8,12] == flat [96]
  const int* src = edges;
  const int* dst = edges + NEDGES;
  float* out = (float*)d_out;

  // Workspace carve-up (all f32): z, c, agg [N*96] each + Wc [256*96]
  float* zbuf = (float*)d_ws;
  float* cbuf = zbuf + (size_t)NNODES * HID;
  float* agg  = cbuf + (size_t)NNODES * HID;
  float* Wc   = agg  + (size_t)NNODES * HID;

  // 1) Repack weights, then WMMA init GEMM -> cbuf holds relu(XW+b)
  repack_w<<<(INPD * HID + 255) / 256, 256, 0, stream>>>(W, Wc);
  dim3 ggrid((NNODES / 16 + 3) / 4, HID / 16);   // (782, 6), 4 waves/block
  init_gemm_wmma<<<ggrid, 128, 0, stream>>>(X, Wc, b, cbuf);

  const int nkBlocks = (NNODES * KCAP + 255) / 256;
  const int eBlocks  = (NEDGES + 255) / 256;
  const int n4       = NNODES * HID / 4;
  const int zBlocks  = (n4 + 255) / 256;

  // 2) 4 routing layers x 6 iterations
  for (int layer = 0; layer < NLAYERS; ++layer) {
    normalize_zc<<<nkBlocks, 256, 0, stream>>>(cbuf, zbuf, cbuf);
    zero_buf4<<<zBlocks, 256, 0, stream>>>((float4*)agg, n4);  // once per layer
    for (int it = 0; it < ROUTIT; ++it) {
      edge_route<<<eBlocks, 256, 0, stream>>>(src, dst, zbuf, cbuf, agg);
      center_update<<<nkBlocks, 256, 0, stream>>>(zbuf, agg, cbuf);
    }
  }

  // 3) Output = relu(final centers)
  relu_out<<<(NNODES * HID + 255) / 256, 256, 0, stream>>>(cbuf, out, NNODES * HID);
}